// PaiNNInteraction_72344429134051
// MI455X (gfx1250) — compile-verified
//
#include <hip/hip_runtime.h>

typedef float v2f __attribute__((ext_vector_type(2)));
typedef float v8f __attribute__((ext_vector_type(8)));

#define F   128
#define F2  256
#define F3  384

// ---------------- block reduction helpers (blockDim == 128) ----------------
__device__ __forceinline__ float blk_reduce_max(float v, float* red, int t) {
    red[t] = v; __syncthreads();
    #pragma unroll
    for (int s = 64; s > 0; s >>= 1) {
        if (t < s) red[t] = fmaxf(red[t], red[t + s]);
        __syncthreads();
    }
    float r = red[0]; __syncthreads();
    return r;
}

__device__ __forceinline__ float blk_reduce_sum(float v, float* red, int t) {
    red[t] = v; __syncthreads();
    #pragma unroll
    for (int s = 64; s > 0; s >>= 1) {
        if (t < s) red[t] = red[t] + red[t + s];
        __syncthreads();
    }
    float r = red[0]; __syncthreads();
    return r;
}

// ---------------- zero-init workspace accumulators ----------------
__global__ void zero_kernel(float* __restrict__ p, size_t n) {
    size_t i = (size_t)blockIdx.x * blockDim.x + threadIdx.x;
    if (i < n) p[i] = 0.0f;
}

// ---------------- pack Wr2 into K-pair float2 layout ----------------
// wr2p[kp][col] = {Wr2[2kp][col], Wr2[2kp+1][col]}  (kp = 0..15, col = 0..16383)
__global__ void pack_wr2_kernel(const float* __restrict__ Wr2, float* __restrict__ wr2p) {
    int idx = blockIdx.x * 256 + threadIdx.x;        // 0 .. 16*16384-1
    if (idx < 16 * 16384) {
        int kp  = idx >> 14;
        int col = idx & 16383;
        v2f val;
        val.x = Wr2[(size_t)(2 * kp) * 16384 + col];
        val.y = Wr2[(size_t)(2 * kp + 1) * 16384 + col];
        *(v2f*)&wr2p[2 * (size_t)idx] = val;
    }
}

// ---------------- kernel A: x = silu(q@W1+b1)@W2 + b2 ----------------
__global__ void ctx_net_kernel(const float* __restrict__ q,
                               const float* __restrict__ W1, const float* __restrict__ b1,
                               const float* __restrict__ W2, const float* __restrict__ b2,
                               float* __restrict__ x_ws) {
    int a = blockIdx.x, t = threadIdx.x;
    __shared__ float qs[F];
    __shared__ float hs[F];
    qs[t] = q[(size_t)a * F + t];
    __syncthreads();
    float acc = b1[t];
    #pragma unroll 4
    for (int k = 0; k < F; ++k) acc += qs[k] * W1[k * F + t];
    hs[t] = acc / (1.0f + __expf(-acc));   // silu
    __syncthreads();
    #pragma unroll
    for (int c3 = 0; c3 < 3; ++c3) {
        int c = c3 * F + t;
        float xa = b2[c];
        #pragma unroll 4
        for (int k = 0; k < F; ++k) xa += hs[k] * W2[k * F3 + c];
        x_ws[(size_t)a * F3 + c] = xa;
    }
}

// ---------------- kernel B: per-pair message + compress + scatter-add ----------------
__global__ void pair_kernel(const float* __restrict__ mu, const float* __restrict__ Wij,
                            const float* __restrict__ dir_ij,
                            const int* __restrict__ idx_i, const int* __restrict__ idx_j,
                            const float* __restrict__ Wc1, const float* __restrict__ bc1,
                            const float* __restrict__ Wc2, const float* __restrict__ bc2,
                            const float* __restrict__ x_ws,
                            float* __restrict__ dq_acc, float* __restrict__ dmu_acc,
                            int n_pairs) {
    int p = blockIdx.x;
    if (p >= n_pairs) return;
    int t = threadIdx.x;
    int j  = idx_j[p];
    int ia = idx_i[p];
    __shared__ float s_lds[F2];
    __shared__ float h1[32];
    __shared__ float red[F];
    const float* muj = mu + (size_t)j * (3 * F2);
    // s[v] = ||mu_j[:, v]||  for 256 channels
    #pragma unroll
    for (int u = 0; u < 2; ++u) {
        int v = t + u * F;
        float x0 = muj[v], x1 = muj[F2 + v], x2 = muj[2 * F2 + v];
        s_lds[v] = sqrtf(x0 * x0 + x1 * x1 + x2 * x2);
    }
    __syncthreads();
    // s @ Wc1 (+bc1), relu : 4 partials of 64 channels each
    {
        int k = t & 31, u = t >> 5;
        float part = 0.0f;
        #pragma unroll 4
        for (int v = u * 64; v < u * 64 + 64; ++v) part += s_lds[v] * Wc1[v * 32 + k];
        red[t] = part;
        __syncthreads();
        if (t < 32) h1[t] = fmaxf(red[t] + red[t + 32] + red[t + 64] + red[t + 96] + bc1[t], 0.0f);
        __syncthreads();
    }
    float logit = bc2[t];
    #pragma unroll
    for (int k = 0; k < 32; ++k) logit += h1[k] * Wc2[k * F + t];
    float mx = blk_reduce_max(logit, red, t);
    float e  = __expf(logit - mx);
    float se = blk_reduce_sum(e, red, t);
    float w  = e / se;                 // softmax weight for channel t
    float vsum[3];
    #pragma unroll
    for (int i = 0; i < 3; ++i) {
        float pv = muj[i * F2 + t] + muj[i * F2 + t + F];
        vsum[i] = blk_reduce_sum(pv, red, t);
    }
    const float* wp = Wij  + (size_t)p * F3;
    const float* xj = x_ws + (size_t)j * F3;
    float dq    = wp[t]         * xj[t];
    float dmuR  = wp[F + t]     * xj[F + t];
    float dmumu = wp[2 * F + t] * xj[2 * F + t];
    atomicAdd(&dq_acc[(size_t)ia * F + t], dq);
    #pragma unroll
    for (int i = 0; i < 3; ++i) {
        float d = dmuR * dir_ij[p * 3 + i] + dmumu * (w * vsum[i]);
        atomicAdd(&dmu_acc[(size_t)ia * F3 + i * F + t], d);
    }
}

// ---------------- kernel C: center-atom compress, combine, h_r = relu(s@Wr1+br1) ----------------
__global__ void atom_combine_kernel(const float* __restrict__ q, const float* __restrict__ mu,
                                    const float* __restrict__ Wc1, const float* __restrict__ bc1,
                                    const float* __restrict__ Wc2, const float* __restrict__ bc2,
                                    const float* __restrict__ Wr1, const float* __restrict__ br1,
                                    const float* __restrict__ dq_acc, const float* __restrict__ dmu_acc,
                                    float* __restrict__ d_out, float* __restrict__ h_ws,
                                    size_t mu_off) {
    int a = blockIdx.x, t = threadIdx.x;
    __shared__ float s_lds[F2];
    __shared__ float h1[32];
    __shared__ float red[F];
    __shared__ float sr[F];
    const float* mua = mu + (size_t)a * (3 * F2);
    #pragma unroll
    for (int u = 0; u < 2; ++u) {
        int v = t + u * F;
        float x0 = mua[v], x1 = mua[F2 + v], x2 = mua[2 * F2 + v];
        s_lds[v] = sqrtf(x0 * x0 + x1 * x1 + x2 * x2);
    }
    __syncthreads();
    {
        int k = t & 31, u = t >> 5;
        float part = 0.0f;
        #pragma unroll 4
        for (int v = u * 64; v < u * 64 + 64; ++v) part += s_lds[v] * Wc1[v * 32 + k];
        red[t] = part;
        __syncthreads();
        if (t < 32) h1[t] = fmaxf(red[t] + red[t + 32] + red[t + 64] + red[t + 96] + bc1[t], 0.0f);
        __syncthreads();
    }
    float logit = bc2[t];
    #pragma unroll
    for (int k = 0; k < 32; ++k) logit += h1[k] * Wc2[k * F + t];
    float mx = blk_reduce_max(logit, red, t);
    float e  = __expf(logit - mx);
    float se = blk_reduce_sum(e, red, t);
    float w  = e / se;
    float vsum[3];
    #pragma unroll
    for (int i = 0; i < 3; ++i) {
        float pv = mua[i * F2 + t] + mua[i * F2 + t + F];
        vsum[i] = blk_reduce_sum(pv, red, t);
    }
    // q_out
    size_t qi = (size_t)a * F + t;
    d_out[qi] = q[qi] + dq_acc[qi];
    // mu_out and s_r
    float ss = 0.0f;
    #pragma unroll
    for (int i = 0; i < 3; ++i) {
        size_t mi = (size_t)a * F3 + i * F + t;
        float mo = dmu_acc[mi] + w * vsum[i];
        d_out[mu_off + mi] = mo;
        ss += mo * mo;
    }
    sr[t] = sqrtf(ss);
    __syncthreads();
    // s_r @ Wr1 (+br1), relu : 4 partials of 32 channels each
    {
        int k = t & 31, u = t >> 5;
        float part = 0.0f;
        #pragma unroll 4
        for (int f2 = u * 32; f2 < u * 32 + 32; ++f2) part += sr[f2] * Wr1[f2 * 32 + k];
        red[t] = part;
        __syncthreads();
        if (t < 32)
            h_ws[(size_t)a * 32 + t] =
                fmaxf(red[t] + red[t + 32] + red[t + 64] + red[t + 96] + br1[t], 0.0f);
    }
}

// ---------------- kernel D: fused WMMA  w = h@Wr2+br2 ; dtm = w @ G(v) ----------------
// 16 atoms / block, 16 waves (512 threads). GEMM1: M=16 atoms, K=32, N=512-chunk of rc.
// GEMM2: per-atom wave, M=16 w-rows, K=32 c-chunk, N=16 (9 valid ij), acc over 4 c-chunks.
#define LW_C  36          // padded c stride (bank-conflict-free A-fragment reads)
#define LW_A  (16 * LW_C) // per-atom stride in lds_w

__global__ void __launch_bounds__(512, 1)
dtm_wmma_kernel(const float* __restrict__ wr2p, const float* __restrict__ br2,
                const float* __restrict__ h_ws, float* __restrict__ d_out,
                int n_atoms, size_t mu_off, size_t dtm_off) {
    __shared__ alignas(16) float lds_w[16 * LW_A];      // 36,864 B  w chunk [atom][r16][c36]
    __shared__ alignas(16) float lds_vv[16 * 384 + 8];  // 24,608 B  v as [atom][c][i]
    __shared__ alignas(16) float lds_h[16 * 32];        //  2,048 B  h_r [atom][k]
    int tid  = threadIdx.x;
    int wave = tid >> 5;
    int lane = tid & 31;
    int a0   = blockIdx.x * 16;
    for (int idx = tid; idx < 16 * 32; idx += 512) {
        int al = idx >> 5;
        int ag = (a0 + al < n_atoms) ? (a0 + al) : (n_atoms - 1);
        lds_h[idx] = h_ws[(size_t)ag * 32 + (idx & 31)];
    }
    for (int idx = tid; idx < 16 * 384; idx += 512) {
        int al = idx / 384, rem = idx % 384;
        int i = rem >> 7, f = rem & 127;
        int ag = (a0 + al < n_atoms) ? (a0 + al) : (n_atoms - 1);
        lds_vv[al * 384 + f * 3 + i] = d_out[mu_off + (size_t)ag * 384 + rem];
    }
    __syncthreads();

    const v2f* wp2 = (const v2f*)wr2p;   // [kp 0..15][col 0..16383]
    int lhalf = lane >> 4;     // 0|1 -> K pair {0,1} or {2,3} of the 4-wide step
    int lmod  = lane & 15;     // M row / N column within the tile
    int myatom = wave;
    int ag_my  = a0 + myatom;
    const float* lw_my = lds_w  + myatom * LW_A;
    const float* vv_my = lds_vv + myatom * 384;
    int n  = lmod;
    int ii = (n < 9) ? (n / 3) : 0;
    int jj = (n < 9) ? (n - 3 * (n / 3)) : 0;
    float msk = (n < 9) ? 1.0f : 0.0f;

    // wave's two GEMM1 N-tile columns (fixed across rblk/cblk up to the chunk base)
    int nloc0 = wave * 16 + lmod;            // tile tt=0 : 0..255
    int nloc1 = nloc0 + 256;                 // tile tt=1 : 256..511
    int r0 = nloc0 >> 5, c0l = nloc0 & 31;
    int r1 = nloc1 >> 5, c1l = nloc1 & 31;

    for (int rblk = 0; rblk < 8; ++rblk) {
        v8f acc2 = {0.f, 0.f, 0.f, 0.f, 0.f, 0.f, 0.f, 0.f};
        for (int cblk = 0; cblk < 4; ++cblk) {
            size_t col0 = (size_t)((rblk * 16 + r0) * 128 + cblk * 32 + c0l);
            size_t col1 = (size_t)((rblk * 16 + r1) * 128 + cblk * 32 + c1l);
            // ---- GEMM1: prefetch fragments, then chained WMMAs ----
            v2f af[8], bf0[8], bf1[8];
            #pragma unroll
            for (int kk = 0; kk < 8; ++kk) {
                int kp = 2 * kk + lhalf;                 // K-pair index
                bf0[kk] = wp2[(size_t)kp * 16384 + col0];
                bf1[kk] = wp2[(size_t)kp * 16384 + col1];
            }
            #pragma unroll
            for (int kk = 0; kk < 8; ++kk)
                af[kk] = *(const v2f*)&lds_h[(lmod << 5) + 4 * kk + 2 * lhalf];
            v8f acc0 = {0.f, 0.f, 0.f, 0.f, 0.f, 0.f, 0.f, 0.f};
            v8f acc1 = {0.f, 0.f, 0.f, 0.f, 0.f, 0.f, 0.f, 0.f};
            #pragma unroll
            for (int kk = 0; kk < 8; ++kk) {
                acc0 = __builtin_amdgcn_wmma_f32_16x16x4_f32(
                    false, af[kk], false, bf0[kk], (short)0, acc0, false, false);
                acc1 = __builtin_amdgcn_wmma_f32_16x16x4_f32(
                    false, af[kk], false, bf1[kk], (short)0, acc1, false, false);
            }
            float bias0 = br2[col0];
            float bias1 = br2[col1];
            #pragma unroll
            for (int v = 0; v < 8; ++v) {
                int am = v + 8 * lhalf;                  // atom row of C tile
                lds_w[am * LW_A + r0 * LW_C + c0l] = acc0[v] + bias0;
                lds_w[am * LW_A + r1 * LW_C + c1l] = acc1[v] + bias1;
            }
            __syncthreads();
            // ---- GEMM2: prefetch fragments, then chained WMMAs ----
            v2f a2[8], b2[8];
            #pragma unroll
            for (int kk = 0; kk < 8; ++kk)
                a2[kk] = *(const v2f*)&lw_my[lmod * LW_C + 4 * kk + 2 * lhalf];
            #pragma unroll
            for (int kk = 0; kk < 8; ++kk) {
                int cg0 = cblk * 32 + 4 * kk + 2 * lhalf;
                b2[kk].x = msk * vv_my[cg0 * 3 + ii] * vv_my[cg0 * 3 + jj];
                b2[kk].y = msk * vv_my[(cg0 + 1) * 3 + ii] * vv_my[(cg0 + 1) * 3 + jj];
            }
            #pragma unroll
            for (int kk = 0; kk < 8; ++kk)
                acc2 = __builtin_amdgcn_wmma_f32_16x16x4_f32(
                    false, a2[kk], false, b2[kk], (short)0, acc2, false, false);
            __syncthreads();
        }
        // store dtm tile: rows rblk*16 .. +15, cols i*3+j
        if (ag_my < n_atoms && lmod < 9) {
            #pragma unroll
            for (int v = 0; v < 8; ++v) {
                int r = rblk * 16 + v + 8 * lhalf;
                d_out[dtm_off + (size_t)ag_my * 1152 + r * 9 + lmod] = acc2[v];
            }
        }
    }
}

// ---------------- launch ----------------
extern "C" void kernel_launch(void* const* d_in, const int* in_sizes, int n_in,
                              void* d_out, int out_size, void* d_ws, size_t ws_size,
                              hipStream_t stream) {
    const float* q      = (const float*)d_in[0];
    const float* mu     = (const float*)d_in[1];
    const float* Wij    = (const float*)d_in[2];
    const float* dir_ij = (const float*)d_in[3];
    const int*   idx_i  = (const int*)d_in[4];
    const int*   idx_j  = (const int*)d_in[5];
    const float* W1  = (const float*)d_in[7];
    const float* b1  = (const float*)d_in[8];
    const float* W2  = (const float*)d_in[9];
    const float* b2  = (const float*)d_in[10];
    const float* Wc1 = (const float*)d_in[11];
    const float* bc1 = (const float*)d_in[12];
    const float* Wc2 = (const float*)d_in[13];
    const float* bc2 = (const float*)d_in[14];
    const float* Wr1 = (const float*)d_in[15];
    const float* br1 = (const float*)d_in[16];
    const float* Wr2 = (const float*)d_in[17];
    const float* br2 = (const float*)d_in[18];

    int n_atoms = in_sizes[0] / F;
    int n_pairs = in_sizes[4];

    float* x_ws    = (float*)d_ws;                        // n_atoms*384
    float* dq_acc  = x_ws    + (size_t)n_atoms * F3;      // n_atoms*128
    float* dmu_acc = dq_acc  + (size_t)n_atoms * F;       // n_atoms*384
    float* h_ws    = dmu_acc + (size_t)n_atoms * F3;      // n_atoms*32
    float* wr2p    = h_ws    + (size_t)n_atoms * 32;      // 2*16*16384 (K-pair packed Wr2)

    size_t mu_off  = (size_t)n_atoms * F;
    size_t dtm_off = mu_off + (size_t)n_atoms * F3;

    size_t nz = (size_t)n_atoms * (F + F3);               // dq_acc + dmu_acc contiguous
    zero_kernel<<<(unsigned)((nz + 255) / 256), 256, 0, stream>>>(dq_acc, nz);
    pack_wr2_kernel<<<(16 * 16384 + 255) / 256, 256, 0, stream>>>(Wr2, wr2p);
    ctx_net_kernel<<<n_atoms, F, 0, stream>>>(q, W1, b1, W2, b2, x_ws);
    pair_kernel<<<n_pairs, F, 0, stream>>>(mu, Wij, dir_ij, idx_i, idx_j,
                                           Wc1, bc1, Wc2, bc2, x_ws,
                                           dq_acc, dmu_acc, n_pairs);
    atom_combine_kernel<<<n_atoms, F, 0, stream>>>(q, mu, Wc1, bc1, Wc2, bc2,
                                                   Wr1, br1, dq_acc, dmu_acc,
                                                   (float*)d_out, h_ws, mu_off);
    int nblk = (n_atoms + 15) / 16;
    dtm_wmma_kernel<<<nblk, 512, 0, stream>>>(wr2p, br2, h_ws, (float*)d_out,
                                              n_atoms, mu_off, dtm_off);
}